// CTCDecoder_1400159339108
// MI455X (gfx1250) — compile-verified
//
#include <hip/hip_runtime.h>
#include <hip/hip_bf16.h>
#include <math.h>

#define TPB    256
#define NWAVES (TPB / 32)
#define UNROLL 5          // 5 x b128 loads in flight per wave per iteration

typedef float v4f __attribute__((ext_vector_type(4)));

// Branchless: keep (bv,bi) as max; strict '>' keeps first occurrence when
// indices are visited in increasing order (per-thread case).
__device__ __forceinline__ void combine_fast(float& bv, int& bi, float v, int i) {
    const bool better = v > bv;
    bv = better ? v : bv;
    bi = better ? i : bi;
}

// Full tie-break (lowest index wins on equal max) for cross-lane merging.
__device__ __forceinline__ void combine_tie(float& bv, int& bi, float v, int i) {
    const bool better = (v > bv) || (v == bv && i < bi);
    bv = better ? v : bv;
    bi = better ? i : bi;
}

// One block per (b,t) row: argmax + max over C contiguous floats.
// Streams probs exactly once with NT-hinted b128 loads (1.45 GB >> 192 MB L2).
__global__ __launch_bounds__(TPB) void ctc_rowmax_kernel(
        const float* __restrict__ probs,
        int* __restrict__ tok,
        float* __restrict__ maxlp,
        int C) {
    const long long row = blockIdx.x;
    const v4f* __restrict__ p = (const v4f*)(probs + row * (long long)C);
    const int n4 = C >> 2;                    // 2750 for C=11000

    float bv = -1.0f;                         // probs >= 0
    int   bi = 0x7fffffff;

    // Uniform-trip batched main loop: SGPR loop counter, no EXEC divergence,
    // UNROLL independent loads issued back-to-back each iteration.
    const int nbatch = n4 / (UNROLL * TPB);   // 2 for C=11000
    int f = threadIdx.x;
    for (int it = 0; it < nbatch; ++it, f += UNROLL * TPB) {
        v4f q[UNROLL];
        #pragma unroll
        for (int u = 0; u < UNROLL; ++u)
            q[u] = __builtin_nontemporal_load(p + f + u * TPB);  // global_load_b128 th:NT
        #pragma unroll
        for (int u = 0; u < UNROLL; ++u) {
            const int base = (f + u * TPB) << 2;
            combine_fast(bv, bi, q[u].x, base + 0);
            combine_fast(bv, bi, q[u].y, base + 1);
            combine_fast(bv, bi, q[u].z, base + 2);
            combine_fast(bv, bi, q[u].w, base + 3);
        }
    }
    // Remainder (190 float4s for C=11000 -> at most one iteration per lane)
    for (; f < n4; f += TPB) {
        const v4f q = __builtin_nontemporal_load(p + f);
        const int base = f << 2;
        combine_fast(bv, bi, q.x, base + 0);
        combine_fast(bv, bi, q.y, base + 1);
        combine_fast(bv, bi, q.z, base + 2);
        combine_fast(bv, bi, q.w, base + 3);
    }

    // wave32 reduction (tie-break toward lowest index, matching jnp.argmax)
    #pragma unroll
    for (int off = 16; off > 0; off >>= 1) {
        const float ov = __shfl_xor(bv, off, 32);
        const int   oi = __shfl_xor(bi, off, 32);
        combine_tie(bv, bi, ov, oi);
    }

    __shared__ float sv[NWAVES];
    __shared__ int   si[NWAVES];
    const int lane = threadIdx.x & 31;
    const int w    = threadIdx.x >> 5;
    if (lane == 0) { sv[w] = bv; si[w] = bi; }
    __syncthreads();

    if (threadIdx.x == 0) {
        #pragma unroll
        for (int k = 1; k < NWAVES; ++k) combine_tie(bv, bi, sv[k], si[k]);
        tok[row]   = bi;
        maxlp[row] = logf(bv + 1e-7f);  // log monotonic: max(log(p+eps)) == log(max(p)+eps)
    }
}

// One block per batch row: pad decoded with -1, score = -sum(maxlp), serial merge.
__global__ __launch_bounds__(TPB) void ctc_merge_kernel(
        const int* __restrict__ tok,
        const float* __restrict__ maxlp,
        float* __restrict__ out,      // [B*T] decoded (as float) then [B] scores
        int B, int T, int blank) {
    const int b = blockIdx.x;
    float* __restrict__ dec = out + (long long)b * T;

    float s = 0.0f;
    for (int t = threadIdx.x; t < T; t += TPB) {
        dec[t] = -1.0f;
        s += maxlp[(long long)b * T + t];
    }

    #pragma unroll
    for (int off = 16; off > 0; off >>= 1) s += __shfl_xor(s, off, 32);
    __shared__ float ss[NWAVES];
    const int lane = threadIdx.x & 31;
    const int w    = threadIdx.x >> 5;
    if (lane == 0) ss[w] = s;
    __syncthreads();   // also guarantees -1 init completes before serial overwrite

    if (threadIdx.x == 0) {
        float tot = 0.0f;
        #pragma unroll
        for (int k = 0; k < NWAVES; ++k) tot += ss[k];
        out[(long long)B * T + b] = -tot;

        // serial merge: drop repeats (vs previous timestep) and blanks, compact
        const int* __restrict__ tb = tok + (long long)b * T;
        int prev = -1, pos = 0;
        for (int t = 0; t < T; ++t) {
            const int tk = tb[t];
            if (tk != prev && tk != blank) dec[pos++] = (float)tk;
            prev = tk;
        }
    }
}

extern "C" void kernel_launch(void* const* d_in, const int* in_sizes, int n_in,
                              void* d_out, int out_size, void* d_ws, size_t ws_size,
                              hipStream_t stream) {
    const float* probs = (const float*)d_in[0];
    const int B = 64, T = 512;
    const int rows = B * T;                              // 32768
    const int C = (int)((long long)in_sizes[0] / rows);  // 11000
    const int blank = C - 1;

    int*   tok = (int*)d_ws;
    float* mlp = (float*)((char*)d_ws + sizeof(int) * (size_t)rows);

    ctc_rowmax_kernel<<<rows, TPB, 0, stream>>>(probs, tok, mlp, C);
    ctc_merge_kernel<<<B, TPB, 0, stream>>>(tok, mlp, (float*)d_out, B, T, blank);
}